// Encoder_6528350290198
// MI455X (gfx1250) — compile-verified
//
#include <hip/hip_runtime.h>

typedef __attribute__((ext_vector_type(2))) float v2f;
typedef __attribute__((ext_vector_type(8))) float v8f;

// ---------------------------------------------------------------------------
// Kernel 1: per-face feature extraction (centroid, unit normal, area) -> (F,8)
// padded to 8 floats per face for aligned float4 traffic; slot 7 = 0.
// ---------------------------------------------------------------------------
__global__ __launch_bounds__(256) void feat_kernel(const float* __restrict__ pos,
                                                   const int* __restrict__ faces,
                                                   float* __restrict__ feat) {
  const int f = blockIdx.x * 256 + threadIdx.x;
  const int i0 = faces[3 * f + 0];
  const int i1 = faces[3 * f + 1];
  const int i2 = faces[3 * f + 2];
  const float v0x = pos[3 * i0 + 0], v0y = pos[3 * i0 + 1], v0z = pos[3 * i0 + 2];
  const float v1x = pos[3 * i1 + 0], v1y = pos[3 * i1 + 1], v1z = pos[3 * i1 + 2];
  const float v2x = pos[3 * i2 + 0], v2y = pos[3 * i2 + 1], v2z = pos[3 * i2 + 2];
  const float e1x = v1x - v0x, e1y = v1y - v0y, e1z = v1z - v0z;
  const float e2x = v2x - v0x, e2y = v2y - v0y, e2z = v2z - v0z;
  const float nx = e1y * e2z - e1z * e2y;
  const float ny = e1z * e2x - e1x * e2z;
  const float nz = e1x * e2y - e1y * e2x;
  const float nrm = sqrtf(nx * nx + ny * ny + nz * nz);
  const float inv = 1.0f / (nrm + 1e-8f);
  float4* o = (float4*)(feat + (size_t)f * 8);
  o[0] = make_float4((v0x + v1x + v2x) * (1.0f / 3.0f),
                     (v0y + v1y + v2y) * (1.0f / 3.0f),
                     (v0z + v1z + v2z) * (1.0f / 3.0f), nx * inv);
  o[1] = make_float4(ny * inv, nz * inv, 0.5f * nrm, 0.0f);
}

// ---------------------------------------------------------------------------
// Fused face_conv (+optional relu + 4:1 loop_pool) using V_WMMA_F32_16X16X4_F32.
//   CIN  : padded input channels (K = 2*CIN combined [self | nbr_mean])
//   CW   : real weight rows (7 for layer 1, else == CIN)
//   HOUT : output channels
//   POOL : relu + pool-by-4 and write (F/4, HOUT); else plain bias add -> (F, HOUT)
// Block = 256 threads = 8 waves; each wave owns a 16-face tile (128 faces/WG).
// K is processed in <=64-wide slabs so static LDS stays < 64KB.
// ---------------------------------------------------------------------------
template <int CIN, int CW, int HOUT, bool POOL>
__global__ __launch_bounds__(256) void conv_kernel(const float* __restrict__ xin,
                                                   const int* __restrict__ nbr,
                                                   const float* __restrict__ Wsm,   // (CW,HOUT)
                                                   const float* __restrict__ Wnm,   // (CW,HOUT)
                                                   const float* __restrict__ bias,  // (HOUT)
                                                   float* __restrict__ out) {
  constexpr int K = 2 * CIN;
  constexpr int KSPLIT = (K > 64) ? 2 : 1;
  constexpr int KH = K / KSPLIT;   // K per slab
  constexpr int KP = KH + 2;       // padded LDS row stride (bank-conflict-free, 8B aligned)
  constexpr int NT = HOUT / 16;    // 16-wide N tiles

  __shared__ float sWT[HOUT * KP]; // W^T slab: sWT[h*KP + k]
  __shared__ float sA[128 * KP];   // A slabs for 8 waves x 16 rows

  const int tid = threadIdx.x;
  const int wave = tid >> 5;
  const int lane = tid & 31;
  const int m = lane & 15;            // N (for B) / M (for A) index within tile
  const int koff = (lane >> 4) * 2;   // K offset held by this half-wave

  const long long fw = (long long)blockIdx.x * 128 + wave * 16; // first face of tile

  // staging identity: 2 lanes per row
  const int row = lane >> 1;
  const int half = lane & 1;
  const long long f = fw + row;
  const int n0 = nbr[3 * f + 0];
  const int n1 = nbr[3 * f + 1];
  const int n2 = nbr[3 * f + 2];
  const float4* xs = (const float4*)(xin + f * CIN);
  const float4* xn0 = (const float4*)(xin + (long long)n0 * CIN);
  const float4* xn1 = (const float4*)(xin + (long long)n1 * CIN);
  const float4* xn2 = (const float4*)(xin + (long long)n2 * CIN);
  float* Arow = &sA[(wave * 16 + row) * KP];

  v8f acc[NT];
#pragma unroll
  for (int n = 0; n < NT; ++n) acc[n] = {};

#pragma unroll
  for (int s = 0; s < KSPLIT; ++s) {
    // ---- stage W^T slab: row h holds combined [Ws;Wn] column h over k-slab ----
    for (int idx = tid; idx < HOUT * KH; idx += 256) {
      const int h = idx / KH;
      const int kl = idx % KH;
      const int kg = s * KH + kl;
      float v = 0.0f;
      if (kg < CIN) {
        if (kg < CW) v = Wsm[kg * HOUT + h];
      } else {
        const int k2 = kg - CIN;
        if (k2 < CW) v = Wnm[k2 * HOUT + h];
      }
      sWT[h * KP + kl] = v;
    }
    // ---- stage A slab: [self | (x[n0]+x[n1]+x[n2])/3] ----
    constexpr int CPL = (KH / 4) / 2; // float4 chunks per lane
#pragma unroll
    for (int c = 0; c < CPL; ++c) {
      const int cc = half * CPL + c;    // local chunk in slab
      const int kg = s * KH + cc * 4;   // global combined-k of chunk start
      float4 v;
      if (kg < CIN) {
        v = xs[kg >> 2];
      } else {
        const int q = (kg - CIN) >> 2;
        const float4 a = xn0[q], b = xn1[q], cg = xn2[q];
        v.x = (a.x + b.x + cg.x) * (1.0f / 3.0f);
        v.y = (a.y + b.y + cg.y) * (1.0f / 3.0f);
        v.z = (a.z + b.z + cg.z) * (1.0f / 3.0f);
        v.w = (a.w + b.w + cg.w) * (1.0f / 3.0f);
      }
      Arow[cc * 4 + 0] = v.x;
      Arow[cc * 4 + 1] = v.y;
      Arow[cc * 4 + 2] = v.z;
      Arow[cc * 4 + 3] = v.w;
    }
    __syncthreads();

    // ---- GEMM slab: D(16xHOUT) += A(16xKH) @ W(KHxHOUT), f32 WMMA ----
    // A layout: lane holds A[m][koff+{0,1}]; B layout: lane holds W^T[n16+m][koff+{0,1}]
    const float* Ab = &sA[(wave * 16 + m) * KP + koff];
    const float* Bb = &sWT[m * KP + koff];
#pragma unroll
    for (int ks = 0; ks < KH / 4; ++ks) {
      const v2f a = *(const v2f*)(Ab + ks * 4);
#pragma unroll
      for (int n = 0; n < NT; ++n) {
        const v2f b = *(const v2f*)(Bb + n * 16 * KP + ks * 4);
        acc[n] = __builtin_amdgcn_wmma_f32_16x16x4_f32(false, a, false, b, (short)0,
                                                       acc[n], false, false);
      }
    }
    __syncthreads();
  }

  // ---- epilogue ----
  if (POOL) {
    // acc VGPR i: lanes0-15 -> face fw+i, lanes16-31 -> face fw+8+i.
    // relu(x+b) then mean over 4 consecutive faces -> pooled rows fw/4 .. fw/4+3.
    const long long pr = (fw >> 2) + (lane >> 4) * 2;
#pragma unroll
    for (int n = 0; n < NT; ++n) {
      const float bv = bias[n * 16 + m];
      float e[8];
#pragma unroll
      for (int i = 0; i < 8; ++i) {
        const float t = acc[n][i] + bv;
        e[i] = t > 0.0f ? t : 0.0f;
      }
      const float p0 = (e[0] + e[1] + e[2] + e[3]) * 0.25f;
      const float p1 = (e[4] + e[5] + e[6] + e[7]) * 0.25f;
      out[pr * HOUT + n * 16 + m] = p0;
      out[(pr + 1) * HOUT + n * 16 + m] = p1;
    }
  } else {
    const float bv = bias[m];
#pragma unroll
    for (int i = 0; i < 8; ++i) {
      const long long r = fw + (lane >> 4) * 8 + i;
      out[r * HOUT + m] = acc[0][i] + bv;
    }
  }
}

// faces0 (int) -> float slot of the output tuple (values < 2^24, exact).
__global__ void copy_faces_kernel(const int* __restrict__ src, float* __restrict__ dst, int n) {
  const int i = blockIdx.x * blockDim.x + threadIdx.x;
  if (i < n) dst[i] = (float)src[i];
}

extern "C" void kernel_launch(void* const* d_in, const int* in_sizes, int n_in,
                              void* d_out, int out_size, void* d_ws, size_t ws_size,
                              hipStream_t stream) {
  const float* pos = (const float*)d_in[0];
  const int* faces3 = (const int*)d_in[1];
  const int* faces0 = (const int*)d_in[4];
  const int* nbr3 = (const int*)d_in[5];
  const int* nbr2 = (const int*)d_in[6];
  const int* nbr1 = (const int*)d_in[7];
  const int* nbr0 = (const int*)d_in[8];
  const float* W1s = (const float*)d_in[9];
  const float* W1n = (const float*)d_in[10];
  const float* b1 = (const float*)d_in[11];
  const float* W2s = (const float*)d_in[12];
  const float* W2n = (const float*)d_in[13];
  const float* b2 = (const float*)d_in[14];
  const float* W3s = (const float*)d_in[15];
  const float* W3n = (const float*)d_in[16];
  const float* b3 = (const float*)d_in[17];
  const float* W4s = (const float*)d_in[18];
  const float* W4n = (const float*)d_in[19];
  const float* b4 = (const float*)d_in[20];

  const int F3 = in_sizes[5] / 3;  // 1048576
  const int F2 = in_sizes[6] / 3;  //  262144
  const int F1 = in_sizes[7] / 3;  //   65536
  const int F0 = in_sizes[8] / 3;  //   16384

  // workspace layout (floats): feat(F3*8) | p3(F2*64) | p2(F1*64) | p1(F0*64)
  float* ws = (float*)d_ws;
  float* feat = ws;
  float* p3 = feat + (size_t)F3 * 8;
  float* p2 = p3 + (size_t)F2 * 64;
  float* p1 = p2 + (size_t)F1 * 64;

  float* outFaces = (float*)d_out;              // F0*3
  float* outX = outFaces + (size_t)F0 * 3;      // F0*16

  feat_kernel<<<F3 / 256, 256, 0, stream>>>(pos, faces3, feat);
  // layer 1: feat(pad 8, real 7) -> relu -> pool -> p3 (F2 x 64)
  conv_kernel<8, 7, 64, true><<<F3 / 128, 256, 0, stream>>>(feat, nbr3, W1s, W1n, b1, p3);
  // layer 2: p3 -> relu -> pool -> p2 (F1 x 64)
  conv_kernel<64, 64, 64, true><<<F2 / 128, 256, 0, stream>>>(p3, nbr2, W2s, W2n, b2, p2);
  // layer 3: p2 -> relu -> pool -> p1 (F0 x 64)
  conv_kernel<64, 64, 64, true><<<F1 / 128, 256, 0, stream>>>(p2, nbr1, W3s, W3n, b3, p1);
  // layer 4: p1 -> (F0 x 16), no relu/pool, straight to output
  conv_kernel<64, 64, 16, false><<<F0 / 128, 256, 0, stream>>>(p1, nbr0, W4s, W4n, b4, outX);
  copy_faces_kernel<<<(F0 * 3 + 255) / 256, 256, 0, stream>>>(faces0, outFaces, F0 * 3);
}